// CustomDenoisingNetwork_19507741458639
// MI455X (gfx1250) — compile-verified
//
#include <hip/hip_runtime.h>

// CustomDenoisingNetwork for MI455X (gfx1250).
// Memory-bound (~66 GFLOP vs ~1.4 GB HBM traffic) -> use exact-precision
// V_WMMA_F32_16X16X4_F32 (fp32 in/out, matches the fp32 reference bitwise-ish)
// and spend all effort minimizing global traffic via LDS-fused GEMM chains.

#define D        128
#define NNODES   100000
#define NEDGES   640000
#define LDA      132   // 132 % 64 == 4 -> bank = 4*row + k, conflict-free for 16 rows
#define LDF      260   // 260 % 64 == 4 -> same property for the 16x256 f1 tile

typedef float v2f __attribute__((ext_vector_type(2)));
typedef float v8f __attribute__((ext_vector_type(8)));

__device__ __forceinline__ v8f splat8(float v) {
  v8f r;
#pragma unroll
  for (int i = 0; i < 8; ++i) r[i] = v;
  return r;
}

// One wave computes a 16x16 fp32 output tile: C += A(16xK, LDS) * W(KxNcols, global).
// A layout (V_WMMA_F32_16X16X4_F32): lane = {row = lane&15, kb = (lane>>4)*2},
//   a.x = A[row][k+kb], a.y = A[row][k+kb+1].
// B layout (symmetric): lane col = lane&15, b.x = W[k+kb][col], b.y = W[k+kb+1][col].
__device__ __forceinline__ v8f wmma_gemm_f32(const float* __restrict__ sA, int lda,
                                             const float* __restrict__ W, int ldw,
                                             int col, int K, v8f c) {
  const int lane = threadIdx.x & 31;
  const int rn   = lane & 15;          // A row == B col for this lane
  const int kb   = (lane >> 4) << 1;   // 0 or 2
  const float* a0 = sA + rn * lda + kb;
  const float* w0 = W + (size_t)kb * ldw + col + rn;
#pragma unroll 8
  for (int k = 0; k < K; k += 4) {
    v2f a, b;
    a.x = a0[k];
    a.y = a0[k + 1];
    b.x = w0[(size_t)k * ldw];
    b.y = w0[(size_t)(k + 1) * ldw];
    c = __builtin_amdgcn_wmma_f32_16x16x4_f32(false, a, false, b, (short)0, c,
                                              false, false);
  }
  return c;
}

// 256 threads load a 16x128 fp32 tile (each thread: 8 contiguous floats, 32B aligned).
__device__ __forceinline__ void load_tile(float* __restrict__ dst,
                                          const float* __restrict__ src,
                                          int rowBase) {
  const int t  = threadIdx.x;
  const int r  = t >> 4;
  const int c0 = (t & 15) * 8;
  const float4* s = (const float4*)(src + (size_t)(rowBase + r) * D + c0);
  float4 v0 = s[0], v1 = s[1];
  float* d = dst + r * LDA + c0;
  *(float4*)(d)     = v0;
  *(float4*)(d + 4) = v1;
}

// ---------------- Kernel 1: Q/K/V projections + h = relu(Q*K) ----------------
__global__ void __launch_bounds__(256) qkv_h_kernel(
    const float* __restrict__ x, const float* __restrict__ ph,
    const float* __restrict__ Wq, const float* __restrict__ bq,
    const float* __restrict__ Wk, const float* __restrict__ bk,
    const float* __restrict__ Wv, const float* __restrict__ bv,
    float* __restrict__ hbuf, float* __restrict__ Vbuf) {
  __shared__ float sx[16 * LDA];
  __shared__ float sp[16 * LDA];
  const int rowBase = blockIdx.x * 16;
  load_tile(sx, x, rowBase);
  load_tile(sp, ph, rowBase);
  __syncthreads();

  const int wave  = threadIdx.x >> 5;
  const int lane  = threadIdx.x & 31;
  const int col   = wave * 16;
  const int n     = lane & 15;
  const int mBase = (lane >> 4) * 8;

  v8f q = splat8(bq[col + n]);
  q = wmma_gemm_f32(sx, LDA, Wq, D, col, D, q);
  v8f kk = splat8(bk[col + n]);
  kk = wmma_gemm_f32(sp, LDA, Wk, D, col, D, kk);
  v8f vv = splat8(bv[col + n]);
  vv = wmma_gemm_f32(sp, LDA, Wv, D, col, D, vv);

#pragma unroll
  for (int j = 0; j < 8; ++j) {
    size_t o = (size_t)(rowBase + mBase + j) * D + col + n;
    float hv = q[j] * kk[j];
    hbuf[o] = hv > 0.f ? hv : 0.f;
    Vbuf[o] = vv[j];
  }
}

// ------------- Kernel 2: fused edge MLP (both GEMMs, e1 stays in LDS) -------------
__global__ void __launch_bounds__(256) edge_kernel(
    const float* __restrict__ ea, const float* __restrict__ pe,
    const float* __restrict__ We1, const float* __restrict__ be1,
    const float* __restrict__ We2, const float* __restrict__ be2,
    float* __restrict__ eout) {
  __shared__ float sA[16 * LDA];   // edge_attr + prompt_e
  __shared__ float sP[16 * LDA];   // prompt_e (read once from HBM, reused)
  __shared__ float sE[16 * LDA];   // relu(e1) + prompt_e
  const int rowBase = blockIdx.x * 16;
  {
    const int t  = threadIdx.x;
    const int r  = t >> 4;
    const int c0 = (t & 15) * 8;
    const float4* p4 = (const float4*)(pe + (size_t)(rowBase + r) * D + c0);
    const float4* a4 = (const float4*)(ea + (size_t)(rowBase + r) * D + c0);
    float4 p0 = p4[0], p1 = p4[1];
    float4 a0 = a4[0], a1 = a4[1];
    float* dp = sP + r * LDA + c0;
    float* da = sA + r * LDA + c0;
    *(float4*)(dp)     = p0;
    *(float4*)(dp + 4) = p1;
    *(float4*)(da)     = make_float4(a0.x + p0.x, a0.y + p0.y, a0.z + p0.z, a0.w + p0.w);
    *(float4*)(da + 4) = make_float4(a1.x + p1.x, a1.y + p1.y, a1.z + p1.z, a1.w + p1.w);
  }
  __syncthreads();

  const int wave  = threadIdx.x >> 5;
  const int lane  = threadIdx.x & 31;
  const int col   = wave * 16;
  const int n     = lane & 15;
  const int mBase = (lane >> 4) * 8;

  v8f c1 = splat8(be1[col + n]);
  c1 = wmma_gemm_f32(sA, LDA, We1, D, col, D, c1);
#pragma unroll
  for (int j = 0; j < 8; ++j) {
    float v = c1[j] > 0.f ? c1[j] : 0.f;
    sE[(mBase + j) * LDA + col + n] = v + sP[(mBase + j) * LDA + col + n];
  }
  __syncthreads();

  v8f c2 = splat8(be2[col + n]);
  c2 = wmma_gemm_f32(sE, LDA, We2, D, col, D, c2);
#pragma unroll
  for (int j = 0; j < 8; ++j) {
    size_t o = (size_t)(rowBase + mBase + j) * D + col + n;
    eout[o] = c2[j] > 0.f ? c2[j] : 0.f;
  }
}

// ---------------- Kernel 3: zero the aggregation buffer ----------------
__global__ void __launch_bounds__(256) zero_kernel(float4* __restrict__ p, int n4) {
  int i = blockIdx.x * 256 + threadIdx.x;
  if (i < n4) p[i] = make_float4(0.f, 0.f, 0.f, 0.f);
}

// -------- Kernel 4: msg = relu(h[src] + e); scatter-add into aggr[dst] --------
__global__ void __launch_bounds__(256) scatter_kernel(
    const int* __restrict__ srcIdx, const int* __restrict__ dstIdx,
    const float* __restrict__ hbuf, const float* __restrict__ ebuf,
    float* __restrict__ aggr) {
  unsigned tid = blockIdx.x * 256u + threadIdx.x;
  unsigned e = tid >> 5;            // 32 threads per edge, 4 floats per thread
  if (e >= NEDGES) return;
  unsigned g = (tid & 31u) * 4u;
  int s = srcIdx[e];
  int d = dstIdx[e];
  float4 hv = *(const float4*)(hbuf + (size_t)s * D + g);   // L2-resident gather
  float4 ev = *(const float4*)(ebuf + (size_t)e * D + g);
  float* ap = aggr + (size_t)d * D + g;
  atomicAdd(ap + 0, fmaxf(hv.x + ev.x, 0.f));
  atomicAdd(ap + 1, fmaxf(hv.y + ev.y, 0.f));
  atomicAdd(ap + 2, fmaxf(hv.z + ev.z, 0.f));
  atomicAdd(ap + 3, fmaxf(hv.w + ev.w, 0.f));
}

// --- Kernel 5: fused Wh + residual + Wf1 + Wf2 (t and f1 never leave LDS) ---
__global__ void __launch_bounds__(256) out_kernel(
    const float* __restrict__ hbuf, const float* __restrict__ Vbuf,
    const float* __restrict__ aggr, const float* __restrict__ x,
    const float* __restrict__ Wh, const float* __restrict__ bh,
    const float* __restrict__ Wf1, const float* __restrict__ bf1,
    const float* __restrict__ Wf2, const float* __restrict__ bf2,
    float* __restrict__ out) {
  __shared__ float sA[16 * LDA];   // h + V + aggr
  __shared__ float sT[16 * LDA];   // t = relu(sA@Wh + bh) + x
  __shared__ float sF[16 * LDF];   // f1 = relu(t@Wf1 + bf1), 16x256
  const int rowBase = blockIdx.x * 16;
  {
    const int t  = threadIdx.x;
    const int r  = t >> 4;
    const int c0 = (t & 15) * 8;
    size_t o = (size_t)(rowBase + r) * D + c0;
    float* dd = sA + r * LDA + c0;
#pragma unroll
    for (int i = 0; i < 8; ++i) dd[i] = hbuf[o + i] + Vbuf[o + i] + aggr[o + i];
  }
  __syncthreads();

  const int wave  = threadIdx.x >> 5;
  const int lane  = threadIdx.x & 31;
  const int col   = wave * 16;
  const int n     = lane & 15;
  const int mBase = (lane >> 4) * 8;

  v8f t1 = splat8(bh[col + n]);
  t1 = wmma_gemm_f32(sA, LDA, Wh, D, col, D, t1);
#pragma unroll
  for (int j = 0; j < 8; ++j) {
    float v = fmaxf(t1[j], 0.f) + x[(size_t)(rowBase + mBase + j) * D + col + n];
    sT[(mBase + j) * LDA + col + n] = v;
  }
  __syncthreads();

  // f1: 256 output cols; each wave covers two 16-wide column tiles.
#pragma unroll
  for (int half = 0; half < 2; ++half) {
    int cf = wave * 32 + half * 16;
    v8f f = splat8(bf1[cf + n]);
    f = wmma_gemm_f32(sT, LDA, Wf1, 2 * D, cf, D, f);
#pragma unroll
    for (int j = 0; j < 8; ++j)
      sF[(mBase + j) * LDF + cf + n] = fmaxf(f[j], 0.f);
  }
  __syncthreads();

  v8f o = splat8(bf2[col + n]);
  o = wmma_gemm_f32(sF, LDF, Wf2, D, col, 2 * D, o);
#pragma unroll
  for (int j = 0; j < 8; ++j)
    out[(size_t)(rowBase + mBase + j) * D + col + n] = o[j];
}

extern "C" void kernel_launch(void* const* d_in, const int* in_sizes, int n_in,
                              void* d_out, int out_size, void* d_ws, size_t ws_size,
                              hipStream_t stream) {
  const float* x   = (const float*)d_in[0];
  const float* ph  = (const float*)d_in[1];
  const float* ea  = (const float*)d_in[2];
  const float* pe  = (const float*)d_in[3];
  const int*   ei  = (const int*)d_in[4];
  const float* Wq  = (const float*)d_in[5];  const float* bq  = (const float*)d_in[6];
  const float* Wk  = (const float*)d_in[7];  const float* bk  = (const float*)d_in[8];
  const float* Wv  = (const float*)d_in[9];  const float* bv  = (const float*)d_in[10];
  const float* We1 = (const float*)d_in[11]; const float* be1 = (const float*)d_in[12];
  const float* We2 = (const float*)d_in[13]; const float* be2 = (const float*)d_in[14];
  const float* Wh  = (const float*)d_in[15]; const float* bh  = (const float*)d_in[16];
  const float* Wf1 = (const float*)d_in[17]; const float* bf1 = (const float*)d_in[18];
  const float* Wf2 = (const float*)d_in[19]; const float* bf2 = (const float*)d_in[20];

  float* out_h = (float*)d_out;                       // [N, D]
  float* out_e = (float*)d_out + (size_t)NNODES * D;  // [E, D]

  float* hbuf = (float*)d_ws;                         // [N, D]
  float* Vbuf = hbuf + (size_t)NNODES * D;            // [N, D]
  float* aggr = Vbuf + (size_t)NNODES * D;            // [N, D]

  const int* srcIdx = ei;           // edge_index[0]
  const int* dstIdx = ei + NEDGES;  // edge_index[1]

  qkv_h_kernel<<<NNODES / 16, 256, 0, stream>>>(x, ph, Wq, bq, Wk, bk, Wv, bv,
                                                hbuf, Vbuf);
  edge_kernel<<<NEDGES / 16, 256, 0, stream>>>(ea, pe, We1, be1, We2, be2, out_e);
  zero_kernel<<<(NNODES * D / 4 + 255) / 256, 256, 0, stream>>>((float4*)aggr,
                                                                NNODES * D / 4);
  scatter_kernel<<<(NEDGES * 32) / 256, 256, 0, stream>>>(srcIdx, dstIdx, hbuf,
                                                          out_e, aggr);
  out_kernel<<<NNODES / 16, 256, 0, stream>>>(hbuf, Vbuf, aggr, x, Wh, bh, Wf1,
                                              bf1, Wf2, bf2, out_h);
}